// TokenCharFusion_30958124269956
// MI455X (gfx1250) — compile-verified
//
#include <hip/hip_runtime.h>
#include <hip/hip_bf16.h>
#include <math.h>

// ---------------------------------------------------------------------------
// TokenCharFusion on MI455X (gfx1250), wave32, WMMA f32 16x16x4,
// double-buffered LDS with async global->LDS weight staging (ASYNCcnt path).
// Sizes fixed by reference: B=16, T=256, C=16, TOKD=1024, D=512, MV=4096.
// ---------------------------------------------------------------------------

typedef __attribute__((ext_vector_type(2))) float v2f;
typedef __attribute__((ext_vector_type(8))) float v8f;

#define BT     4096      // B*T
#define ROWS   65536     // B*T*C
#define TOKD   1024
#define DDIM   512
#define MV     4096
#define LDA    20        // LDS stride (floats): 16 + 4 pad -> 80B rows (16B aligned)
#define LDB    20
#define KC     16        // K chunk per LDS stage
#define ASZ    (64 * LDA)     // floats per A buffer
#define BSZ    (DDIM * LDB)   // floats per B buffer

#if defined(__has_builtin)
#if __has_builtin(__builtin_amdgcn_sched_barrier)
#define SCHED_FENCE() __builtin_amdgcn_sched_barrier(0)
#endif
#endif
#ifndef SCHED_FENCE
#define SCHED_FENCE() asm volatile("" ::: "memory")
#endif

__device__ __forceinline__ float silu_f(float x) {
    return x * (1.0f / (1.0f + __expf(-x)));
}
__device__ __forceinline__ float sigmoid_f(float x) {
    return 1.0f / (1.0f + __expf(-x));
}

__device__ __forceinline__ v8f wmma4(v2f a, v2f b, v8f c) {
    // D = A(16x4) * B(4x16) + C, full fp32
    return __builtin_amdgcn_wmma_f32_16x16x4_f32(
        /*neg_a=*/false, a, /*neg_b=*/false, b,
        /*c_mod=*/(short)0, c, /*reuse_a=*/false, /*reuse_b=*/false);
}

// Async copy of 64 contiguous bytes global -> LDS (one row of a weight tile).
// The offset: field is added to both the LDS and the global address.
__device__ __forceinline__ void async_copy64B(unsigned lds_off, unsigned long long gaddr) {
    asm volatile(
        "global_load_async_to_lds_b128 %0, %1, off\n\t"
        "global_load_async_to_lds_b128 %0, %1, off offset:16\n\t"
        "global_load_async_to_lds_b128 %0, %1, off offset:32\n\t"
        "global_load_async_to_lds_b128 %0, %1, off offset:48"
        :: "v"(lds_off), "v"(gaddr) : "memory");
}
__device__ __forceinline__ void wait_async0() {
    asm volatile("s_wait_asynccnt 0x0" ::: "memory");
}

// ---------------------------------------------------------------------------
// Kernel 1: per-batch stream-compaction scan of the char mask.
// destRow[r] = b*MV + (#valid before r in batch b)  if mask[r], else -1
// grid: <<<16, 256>>>
// ---------------------------------------------------------------------------
__global__ __launch_bounds__(256) void scan_kernel(const unsigned char* __restrict__ mask,
                                                   int* __restrict__ destRow) {
    const int b = blockIdx.x;
    const int t = threadIdx.x;
    __shared__ int s[256];

    const unsigned char* m = mask + b * MV + t * 16;
    unsigned char loc[16];
    int cnt = 0;
#pragma unroll
    for (int i = 0; i < 16; ++i) { loc[i] = m[i]; cnt += loc[i] ? 1 : 0; }

    s[t] = cnt;
    __syncthreads();
    for (int off = 1; off < 256; off <<= 1) {
        int v = (t >= off) ? s[t - off] : 0;
        __syncthreads();
        s[t] += v;
        __syncthreads();
    }
    int run = s[t] - cnt;   // exclusive prefix
    int* d = destRow + b * MV + t * 16;
#pragma unroll
    for (int i = 0; i < 16; ++i) {
        if (loc[i]) { d[i] = b * MV + run; ++run; }
        else        { d[i] = -1; }
    }
}

// ---------------------------------------------------------------------------
// Kernel 2: fill output with broadcast filler_embed. 33.5M floats as float4.
// grid: <<<32768, 256>>>
// ---------------------------------------------------------------------------
__global__ __launch_bounds__(256) void fill_kernel(const float* __restrict__ filler,
                                                   float* __restrict__ out) {
    const long idx = (long)blockIdx.x * 256 + threadIdx.x;    // float4 index
    const int col4 = (int)(idx & 127);
    float4 f = ((const float4*)filler)[col4];
    ((float4*)out)[idx] = f;
}

// ---------------------------------------------------------------------------
// Kernel 3: feats_raw = rmsnorm(token, w_pre) @ W_down^T
//   M=4096 rows, K=1024, N=512. Block = 64 rows x 512 cols, 512 thr (16 waves).
// grid: <<<64, 512>>>
// ---------------------------------------------------------------------------
__global__ __launch_bounds__(512) void gemm_down_kernel(const float* __restrict__ tok,
                                                        const float* __restrict__ w_pre,
                                                        const float* __restrict__ W_down,
                                                        float* __restrict__ feats) {
    __shared__ __align__(16) float As[2][ASZ];
    __shared__ __align__(16) float Bs[2][BSZ];
    __shared__ float red[64][8];
    __shared__ float rstd[64];

    const int t   = threadIdx.x;
    const int rg0 = blockIdx.x * 64;

    // ---- phase 0: row sum-of-squares (8 threads x 128 elems per row) ----
    {
        const int r = t >> 3, seg = t & 7;
        const float* p = tok + (long)(rg0 + r) * TOKD + seg * 128;
        float ss = 0.0f;
#pragma unroll
        for (int i = 0; i < 128; i += 4) {
            float4 v = *(const float4*)(p + i);
            ss += v.x * v.x + v.y * v.y + v.z * v.z + v.w * v.w;
        }
        red[r][seg] = ss;
    }
    __syncthreads();
    if (t < 64) {
        float s = 0.0f;
#pragma unroll
        for (int j = 0; j < 8; ++j) s += red[t][j];
        rstd[t] = rsqrtf(s * (1.0f / (float)TOKD) + 1e-6f);
    }
    __syncthreads();

    const int lane = t & 31, wv = t >> 5;
    const int rt = wv >> 2, ns = wv & 3;
    const int mrow = lane & 15, khalf = lane >> 4;
    const int abase = (rt * 16 + mrow) * LDA + 2 * khalf;

    const unsigned bs_lds0 = (unsigned)(size_t)(&Bs[0][0] + t * LDB);
    const unsigned long long wrow = (unsigned long long)(size_t)(W_down + (long)t * TOKD);
    const int ar = t >> 3, ak = (t & 7) * 2;   // A staging coords

    v8f acc[8];
#pragma unroll
    for (int j = 0; j < 8; ++j) acc[j] = (v8f){0.f,0.f,0.f,0.f,0.f,0.f,0.f,0.f};

    // prologue: stage chunk 0 into buffer 0
    async_copy64B(bs_lds0, wrow);
    {
        float2 x = *(const float2*)(tok + (long)(rg0 + ar) * TOKD + ak);
        float2 w = *(const float2*)(w_pre + ak);
        const float rs = rstd[ar];
        As[0][ar * LDA + ak]     = x.x * rs * w.x;
        As[0][ar * LDA + ak + 1] = x.y * rs * w.y;
    }
    wait_async0();
    __syncthreads();

    const int NC = TOKD / KC;   // 64 chunks
    for (int c = 0; c < NC; ++c) {
        const int cur = c & 1;
        const float* Ab = &As[cur][0];
        const float* Bb = &Bs[cur][0];

        // prefetch next chunk into the other buffer (overlaps the MACs below)
        if (c + 1 < NC) {
            const int k0n = (c + 1) * KC;
            async_copy64B(bs_lds0 + (unsigned)((cur ^ 1) * BSZ * 4), wrow + (unsigned)k0n * 4u);
            float2 x = *(const float2*)(tok + (long)(rg0 + ar) * TOKD + k0n + ak);
            float2 w = *(const float2*)(w_pre + k0n + ak);
            const float rs = rstd[ar];
            As[cur ^ 1][ar * LDA + ak]     = x.x * rs * w.x;
            As[cur ^ 1][ar * LDA + ak + 1] = x.y * rs * w.y;
        }

        // batched fragment loads for the whole chunk, then fenced MAC group
        v2f afr[4];
        v2f bfr[4][8];
#pragma unroll
        for (int q = 0; q < 4; ++q) {
            const int kk = q * 4;
            afr[q] = *(const v2f*)(Ab + abase + kk);
#pragma unroll
            for (int j = 0; j < 8; ++j)
                bfr[q][j] = *(const v2f*)(Bb + (ns * 128 + j * 16 + mrow) * LDB + 2 * khalf + kk);
        }
        SCHED_FENCE();
#pragma unroll
        for (int q = 0; q < 4; ++q)
#pragma unroll
            for (int j = 0; j < 8; ++j)
                acc[j] = wmma4(afr[q], bfr[q][j], acc[j]);

        if (c + 1 < NC) wait_async0();
        __syncthreads();
    }

    // ---- epilogue ----
#pragma unroll
    for (int j = 0; j < 8; ++j) {
        const int n = ns * 128 + j * 16 + mrow;
#pragma unroll
        for (int v = 0; v < 8; ++v) {
            const int m = khalf * 8 + v;
            feats[(long)(rg0 + rt * 16 + m) * DDIM + n] = acc[j][v];
        }
    }
}

// ---------------------------------------------------------------------------
// Kernel 4: per-(b,t) params: rmsnorm(feats,w_tok) -> silu -> @W_proj^T
// grid: <<<4096, 256>>>
// ---------------------------------------------------------------------------
__device__ __forceinline__ float block_reduce256(float v, float* red, int t) {
    __syncthreads();
    red[t] = v;
    __syncthreads();
    for (int off = 128; off > 0; off >>= 1) {
        if (t < off) red[t] += red[t + off];
        __syncthreads();
    }
    return red[0];
}

__global__ __launch_bounds__(256) void params_kernel(const float* __restrict__ feats,
                                                     const float* __restrict__ w_tok,
                                                     const float* __restrict__ W_proj,
                                                     float* __restrict__ smg) {
    const int row = blockIdx.x;
    const int t = threadIdx.x;
    __shared__ float red[256];
    __shared__ float sh_rstd;

    const float v1 = feats[(long)row * DDIM + t];
    const float v2 = feats[(long)row * DDIM + t + 256];
    float total = block_reduce256(v1 * v1 + v2 * v2, red, t);
    if (t == 0) sh_rstd = rsqrtf(total * (1.0f / (float)DDIM) + 1e-6f);
    __syncthreads();
    const float rs = sh_rstd;

    const float s1 = silu_f(v1 * rs * w_tok[t]);
    const float s2 = silu_f(v2 * rs * w_tok[t + 256]);

    float r0 = block_reduce256(s1 * W_proj[t]        + s2 * W_proj[t + 256],        red, t);
    float r1 = block_reduce256(s1 * W_proj[512 + t]  + s2 * W_proj[512 + t + 256],  red, t);
    float r2 = block_reduce256(s1 * W_proj[1024 + t] + s2 * W_proj[1024 + t + 256], red, t);
    if (t == 0) {
        smg[(long)row * 3 + 0] = r0;              // scale
        smg[(long)row * 3 + 1] = r1;              // shift
        smg[(long)row * 3 + 2] = sigmoid_f(r2);   // gate
    }
}

// ---------------------------------------------------------------------------
// Kernel 5 (dominant): fused char path with gate blend + compaction scatter.
//   M=65536, K=512, N=512. Block = 64 rows x 512 cols, 512 thr (16 waves).
// grid: <<<1024, 512>>>
// ---------------------------------------------------------------------------
__global__ __launch_bounds__(512) void gemm_fuse_kernel(const float* __restrict__ chars,
                                                        const float* __restrict__ w_char,
                                                        const float* __restrict__ W_fusion,
                                                        const float* __restrict__ smg,
                                                        const int*   __restrict__ destRow,
                                                        float* __restrict__ out) {
    __shared__ __align__(16) float As[2][ASZ];
    __shared__ __align__(16) float Bs[2][BSZ];
    __shared__ float red[64][8];
    __shared__ float rstd[64], rsc[64], rsh[64], rgt[64];
    __shared__ int   dst[64];
    __shared__ float wch[DDIM];

    const int t   = threadIdx.x;
    const int rg0 = blockIdx.x * 64;   // char-row base

    wch[t] = w_char[t];

    // ---- phase 0: per-row sumsq + per-row params ----
    {
        const int r = t >> 3, seg = t & 7;
        const float* p = chars + (long)(rg0 + r) * DDIM + seg * 64;
        float ss = 0.0f;
#pragma unroll
        for (int i = 0; i < 64; i += 4) {
            float4 v = *(const float4*)(p + i);
            ss += v.x * v.x + v.y * v.y + v.z * v.z + v.w * v.w;
        }
        red[r][seg] = ss;
    }
    __syncthreads();
    if (t < 64) {
        float s = 0.0f;
#pragma unroll
        for (int j = 0; j < 8; ++j) s += red[t][j];
        rstd[t] = rsqrtf(s * (1.0f / (float)DDIM) + 1e-6f);
        const int bt = (rg0 + t) >> 4;          // 16 chars per (b,t)
        rsc[t] = 1.0f + smg[(long)bt * 3 + 0];
        rsh[t] = smg[(long)bt * 3 + 1];
        rgt[t] = smg[(long)bt * 3 + 2];
        dst[t] = destRow[rg0 + t];
    }
    __syncthreads();

    const int lane = t & 31, wv = t >> 5;
    const int rt = wv >> 2, ns = wv & 3;
    const int mrow = lane & 15, khalf = lane >> 4;
    const int abase = (rt * 16 + mrow) * LDA + 2 * khalf;

    const unsigned bs_lds0 = (unsigned)(size_t)(&Bs[0][0] + t * LDB);
    const unsigned long long wrow = (unsigned long long)(size_t)(W_fusion + (long)t * DDIM);
    const int ar = t >> 3, ak = (t & 7) * 2;

    v8f acc[8];
#pragma unroll
    for (int j = 0; j < 8; ++j) acc[j] = (v8f){0.f,0.f,0.f,0.f,0.f,0.f,0.f,0.f};

    // prologue: stage chunk 0 into buffer 0
    async_copy64B(bs_lds0, wrow);
    {
        float2 x = *(const float2*)(chars + (long)(rg0 + ar) * DDIM + ak);
        const float rs = rstd[ar], sc = rsc[ar], sh = rsh[ar];
        As[0][ar * LDA + ak]     = silu_f(x.x * rs * wch[ak] * sc + sh);
        As[0][ar * LDA + ak + 1] = silu_f(x.y * rs * wch[ak + 1] * sc + sh);
    }
    wait_async0();
    __syncthreads();

    const int NC = DDIM / KC;   // 32 chunks
    for (int c = 0; c < NC; ++c) {
        const int cur = c & 1;
        const float* Ab = &As[cur][0];
        const float* Bb = &Bs[cur][0];

        // prefetch next chunk (async weights + modulated A tile)
        if (c + 1 < NC) {
            const int k0n = (c + 1) * KC;
            async_copy64B(bs_lds0 + (unsigned)((cur ^ 1) * BSZ * 4), wrow + (unsigned)k0n * 4u);
            float2 x = *(const float2*)(chars + (long)(rg0 + ar) * DDIM + k0n + ak);
            const float rs = rstd[ar], sc = rsc[ar], sh = rsh[ar];
            As[cur ^ 1][ar * LDA + ak]     = silu_f(x.x * rs * wch[k0n + ak] * sc + sh);
            As[cur ^ 1][ar * LDA + ak + 1] = silu_f(x.y * rs * wch[k0n + ak + 1] * sc + sh);
        }

        v2f afr[4];
        v2f bfr[4][8];
#pragma unroll
        for (int q = 0; q < 4; ++q) {
            const int kk = q * 4;
            afr[q] = *(const v2f*)(Ab + abase + kk);
#pragma unroll
            for (int j = 0; j < 8; ++j)
                bfr[q][j] = *(const v2f*)(Bb + (ns * 128 + j * 16 + mrow) * LDB + 2 * khalf + kk);
        }
        SCHED_FENCE();
#pragma unroll
        for (int q = 0; q < 4; ++q)
#pragma unroll
            for (int j = 0; j < 8; ++j)
                acc[j] = wmma4(afr[q], bfr[q][j], acc[j]);

        if (c + 1 < NC) wait_async0();
        __syncthreads();
    }

    // ---- epilogue: gate blend + compaction scatter ----
#pragma unroll
    for (int j = 0; j < 8; ++j) {
        const int n = ns * 128 + j * 16 + mrow;
        const float wcn = wch[n];
#pragma unroll
        for (int v = 0; v < 8; ++v) {
            const int m = khalf * 8 + v;
            const int r = rt * 16 + m;
            const int d = dst[r];
            if (d >= 0) {
                const float g  = rgt[r];
                const float cn = chars[(long)(rg0 + r) * DDIM + n] * rstd[r] * wcn;
                out[(long)d * DDIM + n] = g * acc[j][v] + (1.0f - g) * cn;
            }
        }
    }
}

// ---------------------------------------------------------------------------
// Host launcher
// ---------------------------------------------------------------------------
extern "C" void kernel_launch(void* const* d_in, const int* in_sizes, int n_in,
                              void* d_out, int out_size, void* d_ws, size_t ws_size,
                              hipStream_t stream) {
    (void)in_sizes; (void)n_in; (void)out_size; (void)ws_size;

    const float*         token    = (const float*)d_in[0];
    // d_in[1] token_ids_mask: unused by the reference computation
    const float*         chars    = (const float*)d_in[2];
    const unsigned char* mask     = (const unsigned char*)d_in[3];
    const float*         filler   = (const float*)d_in[4];
    const float*         w_pre    = (const float*)d_in[5];
    const float*         w_tok    = (const float*)d_in[6];
    const float*         w_char   = (const float*)d_in[7];
    const float*         W_down   = (const float*)d_in[8];
    const float*         W_proj   = (const float*)d_in[9];
    const float*         W_fusion = (const float*)d_in[10];
    float*               out      = (float*)d_out;

    // workspace layout
    float* feats = (float*)d_ws;                 // [4096, 512]  = 8 MB
    float* smg   = feats + (long)BT * DDIM;      // [4096, 3]
    int*   dest  = (int*)(smg + (long)BT * 3);   // [65536]

    scan_kernel  <<<16,    256, 0, stream>>>(mask, dest);
    fill_kernel  <<<32768, 256, 0, stream>>>(filler, out);
    gemm_down_kernel<<<64,  512, 0, stream>>>(token, w_pre, W_down, feats);
    params_kernel<<<BT,    256, 0, stream>>>(feats, w_tok, W_proj, smg);
    gemm_fuse_kernel<<<ROWS / 64, 512, 0, stream>>>(chars, w_char, W_fusion, smg, dest, out);
}